// BotGATEncoder_70858370450099
// MI455X (gfx1250) — compile-verified
//
#include <hip/hip_runtime.h>
#include <math.h>

typedef __attribute__((ext_vector_type(2))) float v2f;
typedef __attribute__((ext_vector_type(8))) float v8f;

#define HID 128

__device__ __forceinline__ float lrelu(float v, float s) { return v > 0.f ? v : s * v; }

// float atomic max via signed-max / unsigned-min bit trick (valid with -inf init)
__device__ __forceinline__ void atomicMaxF(float* addr, float val) {
  if (val >= 0.f) atomicMax((int*)addr, __float_as_int(val));
  else            atomicMin((unsigned int*)addr, __float_as_uint(val));
}

// -------------------- feature encoders (K=7 and K=3, VALU) --------------------
__global__ void encoder_kernel(const float* __restrict__ num_prop, const float* __restrict__ cat_prop,
                               const float* __restrict__ Wn, const float* __restrict__ bn,
                               const float* __restrict__ Wc, const float* __restrict__ bc,
                               float* __restrict__ xcat, int N) {
  int t = blockIdx.x * blockDim.x + threadIdx.x;
  if (t >= N * 64) return;
  int n = t >> 6, j = t & 63;
  const float* np = num_prop + (size_t)n * 7;
  float an = bn[j];
#pragma unroll
  for (int k = 0; k < 7; ++k) an += np[k] * Wn[k * 64 + j];
  const float* cp = cat_prop + (size_t)n * 3;
  float ac = bc[j];
#pragma unroll
  for (int k = 0; k < 3; ++k) ac += cp[k] * Wc[k * 64 + j];
  xcat[(size_t)n * HID + j]      = lrelu(an, 0.01f);
  xcat[(size_t)n * HID + 64 + j] = lrelu(ac, 0.01f);
}

// -------------------- fp32 WMMA GEMM: C[M,128] = act(A[M,128] @ B[128,128] + bias) ----
// One wave -> one 16x16 C tile via V_WMMA_F32_16X16X4_F32, K swept in steps of 4.
// Block = 256 threads = 8 waves -> 16 rows x 128 cols per block.
__global__ void wmma_gemm128_kernel(const float* __restrict__ A, const float* __restrict__ B,
                                    const float* __restrict__ bias, float* __restrict__ C,
                                    int M, int act, int has_bias) {
  int wave = threadIdx.x >> 5;
  int lane = threadIdx.x & 31;
  int row0 = blockIdx.x * 16;
  int col0 = wave * 16;
  int half = lane >> 4;     // 0: lanes 0-15, 1: lanes 16-31
  int m16  = lane & 15;

  int arow = row0 + m16; if (arow >= M) arow = M - 1;   // clamp (EXEC must stay full)
  const float* Ap = A + (size_t)arow * HID + 2 * half;              // A[m][k0+2*half .. +1]
  const float* Bp = B + (size_t)(2 * half) * HID + col0 + m16;      // B[k0+2*half][col]

  v8f acc = {0.f, 0.f, 0.f, 0.f, 0.f, 0.f, 0.f, 0.f};
#pragma unroll 8
  for (int k0 = 0; k0 < HID; k0 += 4) {
    v2f a, b;
    a.x = Ap[k0];
    a.y = Ap[k0 + 1];
    b.x = Bp[(size_t)k0 * HID];
    b.y = Bp[(size_t)(k0 + 1) * HID];
    acc = __builtin_amdgcn_wmma_f32_16x16x4_f32(false, a, false, b, (short)0, acc, false, false);
  }

  int col = col0 + m16;
  float bv = has_bias ? bias[col] : 0.f;
#pragma unroll
  for (int r = 0; r < 8; ++r) {
    int row = row0 + r + 8 * half;   // D layout: VGPR r -> M = r + 8*half
    if (row < M) {
      float v = acc[r] + bv;
      if (act) v = lrelu(v, 0.01f);
      C[(size_t)row * HID + col] = v;
    }
  }
}

// -------------------- BatchNorm (batch statistics) --------------------
__global__ void bn_stats_kernel(const float* __restrict__ x, float* __restrict__ stats, int N) {
  int c = threadIdx.x;  // 128 threads
  float s = 0.f, s2 = 0.f;
  for (int n = blockIdx.x; n < N; n += gridDim.x) {
    float v = x[(size_t)n * HID + c];
    s += v; s2 += v * v;
  }
  atomicAdd(stats + c, s);
  atomicAdd(stats + HID + c, s2);
}

__global__ void bn_finalize_kernel(float* __restrict__ stats, const float* __restrict__ gamma,
                                   const float* __restrict__ beta, int N) {
  int c = threadIdx.x;
  float inv_n = 1.f / (float)N;
  float mu  = stats[c] * inv_n;
  float var = stats[HID + c] * inv_n - mu * mu;
  float sc  = gamma[c] * rsqrtf(var + 1e-5f);
  stats[2 * HID + c] = sc;
  stats[3 * HID + c] = beta[c] - mu * sc;
}

__global__ void bn_apply_kernel(const float* __restrict__ x, const float* __restrict__ stats,
                                float* __restrict__ y, int N) {
  int t = blockIdx.x * blockDim.x + threadIdx.x;
  if (t >= N * HID) return;
  int c = t & (HID - 1);
  y[t] = x[t] * stats[2 * HID + c] + stats[3 * HID + c];
}

// -------------------- attention logits: als/ald[n,h] = <h[n,h,:], a_src/dst[h,:]> -----
__global__ void att_logits_kernel(const float* __restrict__ h, const float* __restrict__ a_src,
                                  const float* __restrict__ a_dst, float* __restrict__ als,
                                  float* __restrict__ ald, int N, int H, int F) {
  int t = blockIdx.x * blockDim.x + threadIdx.x;
  if (t >= N * H) return;
  int n = t / H, hh = t - n * H;
  const float* hp = h + (size_t)n * HID + hh * F;
  const float* as = a_src + hh * F;
  const float* ad = a_dst + hh * F;
  float ss = 0.f, sd = 0.f;
  for (int f = 0; f < F; ++f) { float v = hp[f]; ss += v * as[f]; sd += v * ad[f]; }
  als[t] = ss;
  ald[t] = sd;
}

// -------------------- generic fill --------------------
__global__ void fill_kernel(float* __restrict__ p, long long n, float v) {
  long long t = (long long)blockIdx.x * blockDim.x + threadIdx.x;
  if (t < n) p[t] = v;
}

__device__ __forceinline__ void edge_sd(const int* __restrict__ ei, int E, long long e,
                                        int& s, int& d) {
  if (e < E) { s = ei[e]; d = ei[(size_t)E + e]; }
  else       { s = d = (int)(e - E); }   // self-loop
}

// -------------------- edge pass 1: segment max --------------------
__global__ void edge_max_kernel(const int* __restrict__ ei, const float* __restrict__ als,
                                const float* __restrict__ ald, float* __restrict__ emax,
                                int E, int N, int H) {
  long long t = (long long)blockIdx.x * blockDim.x + threadIdx.x;
  long long total = (long long)(E + N) * H;
  if (t >= total) return;
  long long e = t / H; int hh = (int)(t - e * H);
  int s, d; edge_sd(ei, E, e, s, d);
  float v = lrelu(als[(size_t)s * H + hh] + ald[(size_t)d * H + hh], 0.2f);
  atomicMaxF(emax + (size_t)d * H + hh, v);
}

// -------------------- edge pass 2: denom = segment sum of exp --------------------
__global__ void edge_denom_kernel(const int* __restrict__ ei, const float* __restrict__ als,
                                  const float* __restrict__ ald, const float* __restrict__ emax,
                                  float* __restrict__ denom, int E, int N, int H) {
  long long t = (long long)blockIdx.x * blockDim.x + threadIdx.x;
  long long total = (long long)(E + N) * H;
  if (t >= total) return;
  long long e = t / H; int hh = (int)(t - e * H);
  int s, d; edge_sd(ei, E, e, s, d);
  float v = lrelu(als[(size_t)s * H + hh] + ald[(size_t)d * H + hh], 0.2f);
  float ee = __expf(v - emax[(size_t)d * H + hh]);
  atomicAdd(denom + (size_t)d * H + hh, ee);
}

// -------------------- edge pass 3: scatter alpha * h[src] into acc[dst] --------------
// One wave32 per edge; lane covers cols {lane, lane+32, lane+64, lane+96};
// alpha computed once per head in lanes 0..H-1 and broadcast via shuffle.
__global__ void edge_scatter_kernel(const int* __restrict__ ei, const float* __restrict__ hfeat,
                                    const float* __restrict__ als, const float* __restrict__ ald,
                                    const float* __restrict__ emax, const float* __restrict__ denom,
                                    float* __restrict__ acc, int E, int N, int H, int F) {
  long long wid = ((long long)blockIdx.x * blockDim.x + threadIdx.x) >> 5;
  int lane = threadIdx.x & 31;
  long long total = (long long)E + N;
  if (wid >= total) return;
  int s, d; edge_sd(ei, E, wid, s, d);
  float alpha_l = 0.f;
  if (lane < H) {
    float ev = lrelu(als[(size_t)s * H + lane] + ald[(size_t)d * H + lane], 0.2f);
    alpha_l = __expf(ev - emax[(size_t)d * H + lane]) / denom[(size_t)d * H + lane];
  }
  const float* hs = hfeat + (size_t)s * HID;
  float* ap = acc + (size_t)d * HID;
#pragma unroll
  for (int i = 0; i < 4; ++i) {
    int c = lane + 32 * i;
    int head = c / F;
    float alpha = __shfl(alpha_l, head, 32);
    atomicAdd(ap + c, alpha * hs[c]);
  }
}

// -------------------- add bias (+ optional lrelu), in place --------------------
__global__ void finalize_kernel(float* __restrict__ acc, const float* __restrict__ bias,
                                int N, int act) {
  int t = blockIdx.x * blockDim.x + threadIdx.x;
  if (t >= N * HID) return;
  int c = t & (HID - 1);
  float v = acc[t] + bias[c];
  if (act) v = lrelu(v, 0.01f);
  acc[t] = v;
}

extern "C" void kernel_launch(void* const* d_in, const int* in_sizes, int n_in,
                              void* d_out, int out_size, void* d_ws, size_t ws_size,
                              hipStream_t stream) {
  const float* num_prop = (const float*)d_in[0];
  const float* cat_prop = (const float*)d_in[1];
  const int*   ei       = (const int*)d_in[2];
  const float* W_num = (const float*)d_in[3];
  const float* b_num = (const float*)d_in[4];
  const float* W_cat = (const float*)d_in[5];
  const float* b_cat = (const float*)d_in[6];
  const float* W_in  = (const float*)d_in[7];
  const float* b_in  = (const float*)d_in[8];
  const float* bn_g  = (const float*)d_in[9];
  const float* bn_b  = (const float*)d_in[10];
  const float* W1    = (const float*)d_in[11];
  const float* a_s1  = (const float*)d_in[12];
  const float* a_d1  = (const float*)d_in[13];
  const float* bias1 = (const float*)d_in[14];
  const float* W2    = (const float*)d_in[15];
  const float* a_s2  = (const float*)d_in[16];
  const float* a_d2  = (const float*)d_in[17];
  const float* bias2 = (const float*)d_in[18];

  const int N = in_sizes[0] / 7;
  const int E = in_sizes[2] / 2;
  float* out = (float*)d_out;

  // workspace layout
  float* ws   = (float*)d_ws;
  float* xcat = ws;                       // N*128 (reused as xbn)
  float* x1   = xcat + (size_t)N * HID;   // N*128
  float* hbuf = x1   + (size_t)N * HID;   // N*128
  float* accb = hbuf + (size_t)N * HID;   // N*128 (layer1 acc -> x2)
  float* als  = accb + (size_t)N * HID;   // N*4
  float* ald  = als  + (size_t)N * 4;     // N*4
  float* emax = ald  + (size_t)N * 4;     // N*4
  float* denom= emax + (size_t)N * 4;     // N*4
  float* stats= denom+ (size_t)N * 4;     // 512

  const int TB = 256;
  const long long EN = (long long)E + N;
  const int mtiles = (N + 15) / 16;
  auto nb = [&](long long work) { return (unsigned)((work + TB - 1) / TB); };

  // encoder + input MLP
  encoder_kernel<<<nb((long long)N * 64), TB, 0, stream>>>(num_prop, cat_prop, W_num, b_num,
                                                           W_cat, b_cat, xcat, N);
  wmma_gemm128_kernel<<<mtiles, TB, 0, stream>>>(xcat, W_in, b_in, x1, N, 1, 1);

  // batchnorm
  fill_kernel<<<1, 256, 0, stream>>>(stats, 256, 0.f);
  bn_stats_kernel<<<512, 128, 0, stream>>>(x1, stats, N);
  bn_finalize_kernel<<<1, 128, 0, stream>>>(stats, bn_g, bn_b, N);
  bn_apply_kernel<<<nb((long long)N * HID), TB, 0, stream>>>(x1, stats, xcat, N);

  // ---- GAT layer 1 (H=4, F=32) ----
  wmma_gemm128_kernel<<<mtiles, TB, 0, stream>>>(xcat, W1, nullptr, hbuf, N, 0, 0);
  att_logits_kernel<<<nb((long long)N * 4), TB, 0, stream>>>(hbuf, a_s1, a_d1, als, ald, N, 4, 32);
  fill_kernel<<<nb((long long)N * 4), TB, 0, stream>>>(emax, (long long)N * 4, -INFINITY);
  fill_kernel<<<nb((long long)N * 4), TB, 0, stream>>>(denom, (long long)N * 4, 0.f);
  fill_kernel<<<nb((long long)N * HID), TB, 0, stream>>>(accb, (long long)N * HID, 0.f);
  edge_max_kernel<<<nb(EN * 4), TB, 0, stream>>>(ei, als, ald, emax, E, N, 4);
  edge_denom_kernel<<<nb(EN * 4), TB, 0, stream>>>(ei, als, ald, emax, denom, E, N, 4);
  edge_scatter_kernel<<<nb(EN * 32), TB, 0, stream>>>(ei, hbuf, als, ald, emax, denom, accb,
                                                      E, N, 4, 32);
  finalize_kernel<<<nb((long long)N * HID), TB, 0, stream>>>(accb, bias1, N, 1);

  // ---- GAT layer 2 (H=1, F=128) ----
  wmma_gemm128_kernel<<<mtiles, TB, 0, stream>>>(accb, W2, nullptr, hbuf, N, 0, 0);
  att_logits_kernel<<<nb(N), TB, 0, stream>>>(hbuf, a_s2, a_d2, als, ald, N, 1, 128);
  fill_kernel<<<nb(N), TB, 0, stream>>>(emax, N, -INFINITY);
  fill_kernel<<<nb(N), TB, 0, stream>>>(denom, N, 0.f);
  fill_kernel<<<nb((long long)N * HID), TB, 0, stream>>>(out, (long long)N * HID, 0.f);
  edge_max_kernel<<<nb(EN), TB, 0, stream>>>(ei, als, ald, emax, E, N, 1);
  edge_denom_kernel<<<nb(EN), TB, 0, stream>>>(ei, als, ald, emax, denom, E, N, 1);
  edge_scatter_kernel<<<nb(EN * 32), TB, 0, stream>>>(ei, hbuf, als, ald, emax, denom, out,
                                                      E, N, 1, 128);
  finalize_kernel<<<nb((long long)N * HID), TB, 0, stream>>>(out, bias2, N, 0);
}